// FAPELoss_30932354466006
// MI455X (gfx1250) — compile-verified
//
#include <hip/hip_runtime.h>

typedef __attribute__((ext_vector_type(2))) float v2f;
typedef __attribute__((ext_vector_type(8))) float v8f;

constexpr int   BB      = 4;
constexpr int   NN      = 2048;
constexpr int   TILES   = (NN / 16) * (NN / 16);   // 16384 tiles per batch
constexpr int   NWAVES  = BB * TILES;              // 65536 waves total
constexpr float FEPS    = 1e-4f;
constexpr float FCLAMP  = 10.0f;
constexpr float FSCALE  = 1.0f / (10.0f * (float)NN * (float)NN); // (1/Z) * mean

// Workspace layout (in floats):
//   A: [B][3][N][8]  factorized row data  (196608 floats)
//   V: [B][N][8]     factorized col data  ( 65536 floats)
//   P: [NWAVES]      per-wave partials    ( 65536 floats)
// total = 327680 floats = 1.25 MB (fully L2-resident)
constexpr size_t A_OFF = 0;
constexpr size_t V_OFF = (size_t)BB * 3 * NN * 8;
constexpr size_t P_OFF = V_OFF + (size_t)BB * NN * 8;

// ---------------------------------------------------------------------------
// Phase 1: per (b,i) build the rank-8 factorization.
//   u[i,o] = [RpT[o,0..2], -RtT[o,0..2], tp_inv[o]-tt_inv[o], 0]
//   v[j]   = [xp[j,0..2],  xt[j,0..2],  1,                    0]
// so that d[i,j,o] = u[i,o] . v[j]  (K=8 GEMM per output component)
// ---------------------------------------------------------------------------
__global__ void fape_build(const float* __restrict__ Rp, const float* __restrict__ tp,
                           const float* __restrict__ xp, const float* __restrict__ Rt,
                           const float* __restrict__ tt, const float* __restrict__ xt,
                           float* __restrict__ ws) {
  int t = blockIdx.x * blockDim.x + threadIdx.x;   // [0, B*N)
  if (t >= BB * NN) return;
  int b = t / NN, i = t % NN;

  const float* rp = Rp + (size_t)t * 9;
  const float* rt = Rt + (size_t)t * 9;
  float tpv0 = tp[t * 3 + 0], tpv1 = tp[t * 3 + 1], tpv2 = tp[t * 3 + 2];
  float ttv0 = tt[t * 3 + 0], ttv1 = tt[t * 3 + 1], ttv2 = tt[t * 3 + 2];

  float* A = ws + A_OFF;
  float* V = ws + V_OFF;

#pragma unroll
  for (int o = 0; o < 3; ++o) {
    // R_inv = R^T, so row o of R_inv is column o of R.
    float rpo0 = rp[0 * 3 + o], rpo1 = rp[1 * 3 + o], rpo2 = rp[2 * 3 + o];
    float rto0 = rt[0 * 3 + o], rto1 = rt[1 * 3 + o], rto2 = rt[2 * 3 + o];
    float tpi = -(rpo0 * tpv0 + rpo1 * tpv1 + rpo2 * tpv2);
    float tti = -(rto0 * ttv0 + rto1 * ttv1 + rto2 * ttv2);
    float* a = A + (((size_t)(b * 3 + o)) * NN + i) * 8;
    a[0] = rpo0;  a[1] = rpo1;  a[2] = rpo2;
    a[3] = -rto0; a[4] = -rto1; a[5] = -rto2;
    a[6] = tpi - tti; a[7] = 0.0f;
  }

  float* v = V + ((size_t)b * NN + i) * 8;
  v[0] = xp[t * 3 + 0]; v[1] = xp[t * 3 + 1]; v[2] = xp[t * 3 + 2];
  v[3] = xt[t * 3 + 0]; v[4] = xt[t * 3 + 1]; v[5] = xt[t * 3 + 2];
  v[6] = 1.0f;          v[7] = 0.0f;
}

// ---------------------------------------------------------------------------
// Phase 2: one wave32 per 16x16 pair tile. 3 components x (K=8 as 2 x K=4)
// = 6 x V_WMMA_F32_16X16X4_F32, then hw-sqrt/clamp epilogue + wave reduction.
// ---------------------------------------------------------------------------
__global__ void fape_tile(const float* __restrict__ ws, float* __restrict__ partials) {
  const float* A = ws + A_OFF;
  const float* V = ws + V_OFF;

  int wave = blockIdx.x * (blockDim.x >> 5) + (threadIdx.x >> 5);
  int lane = threadIdx.x & 31;
  int b    = wave / TILES;
  int tile = wave % TILES;
  int i0   = (tile >> 7) << 4;   // 128 tiles per row of the NxN pair space
  int j0   = (tile & 127) << 4;

  int mn = lane & 15;            // M index (A) / N index (B)
  int kb = (lane >> 4) << 1;     // K sub-base: 0 for lanes 0-15, 2 for 16-31

  // B operand (V tile): shared across all 3 components. 8B-aligned float2.
  const float* vrow = V + ((size_t)b * NN + (j0 + mn)) * 8;
  v2f bv0 = *(const v2f*)(vrow + kb);       // K = 0..3 chunk
  v2f bv1 = *(const v2f*)(vrow + 4 + kb);   // K = 4..7 chunk

  v8f acc[3];
#pragma unroll
  for (int o = 0; o < 3; ++o) {
    const float* ar = A + (((size_t)(b * 3 + o)) * NN + (i0 + mn)) * 8;
    v2f a0 = *(const v2f*)(ar + kb);
    v2f a1 = *(const v2f*)(ar + 4 + kb);
    v8f c = {};
    c = __builtin_amdgcn_wmma_f32_16x16x4_f32(false, a0, false, bv0,
                                              (short)0, c, false, false);
    c = __builtin_amdgcn_wmma_f32_16x16x4_f32(false, a1, false, bv1,
                                              (short)0, c, false, false);
    acc[o] = c;
  }

  // Epilogue: each lane holds 8 tile elements; dist = min(sqrt(|d|^2+eps), 10).
  // Raw v_sqrt_f32 (~1 ulp) — argument is >= 1e-4, far from the denormal
  // range the libm fixup sequence exists to handle.
  float local = 0.0f;
#pragma unroll
  for (int r = 0; r < 8; ++r) {
    float s = acc[0][r] * acc[0][r] + acc[1][r] * acc[1][r] + acc[2][r] * acc[2][r];
    float d = __builtin_amdgcn_sqrtf(s + FEPS);
    local += fminf(d, FCLAMP);
  }

  // Wave32 butterfly reduction.
#pragma unroll
  for (int off = 16; off >= 1; off >>= 1)
    local += __shfl_xor(local, off, 32);

  if (lane == 0) partials[wave] = local;
}

// ---------------------------------------------------------------------------
// Phase 3: deterministic tree reduction of per-wave partials (one block per
// batch; fixed accumulation order => bitwise-stable across graph replays).
// ---------------------------------------------------------------------------
__global__ void fape_reduce(const float* __restrict__ partials, float* __restrict__ out) {
  __shared__ float sm[256];
  int b = blockIdx.x;
  float s = 0.0f;
  for (int k = threadIdx.x; k < TILES; k += 256)
    s += partials[(size_t)b * TILES + k];
  sm[threadIdx.x] = s;
  __syncthreads();
  for (int stride = 128; stride >= 1; stride >>= 1) {
    if ((int)threadIdx.x < stride) sm[threadIdx.x] += sm[threadIdx.x + stride];
    __syncthreads();
  }
  if (threadIdx.x == 0) out[b] = sm[0] * FSCALE;
}

extern "C" void kernel_launch(void* const* d_in, const int* in_sizes, int n_in,
                              void* d_out, int out_size, void* d_ws, size_t ws_size,
                              hipStream_t stream) {
  const float* Rp = (const float*)d_in[0];
  const float* tp = (const float*)d_in[1];
  const float* xp = (const float*)d_in[2];
  const float* Rt = (const float*)d_in[3];
  const float* tt = (const float*)d_in[4];
  const float* xt = (const float*)d_in[5];
  float* ws = (float*)d_ws;

  fape_build<<<(BB * NN + 255) / 256, 256, 0, stream>>>(Rp, tp, xp, Rt, tt, xt, ws);
  // 8 waves per 256-thread block, one 16x16 tile per wave.
  fape_tile<<<NWAVES / 8, 256, 0, stream>>>(ws, ws + P_OFF);
  fape_reduce<<<BB, 256, 0, stream>>>(ws + P_OFF, (float*)d_out);
}